// PlainGCN_72919954751682
// MI455X (gfx1250) — compile-verified
//
#include <hip/hip_runtime.h>

// GCN propagate: out[t] += (deg[t]^-1/2 * deg[h]^-1/2) * x[h]; relu.
// N=100000 nodes, E=600000 edges, D=128 fp32 features.
// Memory/atomic bound (AI ~0.25 FLOP/B) -> no WMMA; use CDNA5 async
// global->LDS double-buffered gather + hardware f32 global atomics.
// (Scatter kernel placed first in the file so the disasm snippet shows it.)

#define D_FEAT 128
#define WPB 4            // waves per block in scatter kernel (128 threads)
#define SCATTER_BLOCKS 2048

// One wave32 per edge. Lane L owns float4 chunk L of the 128-float row.
// Double-buffered async gather of x[h[e]] rows into LDS (ASYNCcnt pipeline),
// consume with f32 global atomics into out[t[e]].
__global__ __launch_bounds__(128) void gcn_scatter_kernel(const float* __restrict__ x,
                                                          const int* __restrict__ h,
                                                          const int* __restrict__ t,
                                                          const float* __restrict__ dinv,
                                                          float* __restrict__ out,
                                                          int n_edges) {
    __shared__ float4 stage[WPB][2][32];
    const int lane = threadIdx.x & 31;
    const int wave = threadIdx.x >> 5;
    const int wid  = blockIdx.x * WPB + wave;
    const int nw   = gridDim.x * WPB;

    // LDS byte offsets for this lane's slot in each buffer (flat addr low 32 bits).
    unsigned lds0 = (unsigned)(unsigned long long)(void*)&stage[wave][0][lane];
    unsigned lds1 = (unsigned)(unsigned long long)(void*)&stage[wave][1][lane];

    int e = wid;
    if (e >= n_edges) return;   // wave-uniform exit: EXEC stays all-ones below

    // Prologue: async gather row x[h[e]] -> buffer 0 (one instr per wave, 512B).
    {
        unsigned long long src =
            (unsigned long long)(x + (long long)h[e] * D_FEAT + lane * 4);
        asm volatile("global_load_async_to_lds_b128 %0, %1, off"
                     :: "v"(lds0), "v"(src) : "memory");
    }

    int buf = 0;
    for (;;) {
        int  en   = e + nw;
        bool more = en < n_edges;   // wave-uniform
        if (more) {
            // Prefetch next edge's row into the other buffer, then wait for current.
            unsigned long long src =
                (unsigned long long)(x + (long long)h[en] * D_FEAT + lane * 4);
            unsigned dst = (buf == 0) ? lds1 : lds0;
            asm volatile("global_load_async_to_lds_b128 %0, %1, off"
                         :: "v"(dst), "v"(src) : "memory");
            asm volatile("s_wait_asynccnt 1" ::: "memory");
        } else {
            asm volatile("s_wait_asynccnt 0" ::: "memory");
        }

        // Consume current buffer.
        int   ti  = t[e];
        float nrm = dinv[ti] * dinv[h[e]];
        float4 v  = stage[wave][buf][lane];     // ds_load_b128
        float* orow = out + (long long)ti * D_FEAT + lane * 4;
        unsafeAtomicAdd(orow + 0, nrm * v.x);   // global_atomic_add_f32 (no return)
        unsafeAtomicAdd(orow + 1, nrm * v.y);
        unsafeAtomicAdd(orow + 2, nrm * v.z);
        unsafeAtomicAdd(orow + 3, nrm * v.w);

        if (!more) break;
        e = en;
        buf ^= 1;
    }
}

__global__ __launch_bounds__(256) void gcn_zero_kernel(float4* __restrict__ out4, int n4,
                                                       int* __restrict__ deg, int n_nodes) {
    int i = blockIdx.x * blockDim.x + threadIdx.x;
    if (i < n4) out4[i] = make_float4(0.f, 0.f, 0.f, 0.f);
    if (i < n_nodes) deg[i] = 0;
}

__global__ __launch_bounds__(256) void gcn_degree_kernel(const int* __restrict__ h,
                                                         int* __restrict__ deg, int n_edges) {
    int e = blockIdx.x * blockDim.x + threadIdx.x;
    if (e < n_edges) atomicAdd(&deg[h[e]], 1);   // global_atomic_add_u32 (no return)
}

__global__ __launch_bounds__(256) void gcn_dinv_kernel(const int* __restrict__ deg,
                                                       float* __restrict__ dinv, int n_nodes) {
    int i = blockIdx.x * blockDim.x + threadIdx.x;
    if (i < n_nodes) dinv[i] = rsqrtf((float)deg[i]);   // deg >= 1 by construction
}

__global__ __launch_bounds__(256) void gcn_relu_kernel(float4* __restrict__ out4, int n4) {
    int i = blockIdx.x * blockDim.x + threadIdx.x;
    if (i < n4) {
        float4 v = out4[i];
        v.x = fmaxf(v.x, 0.f);
        v.y = fmaxf(v.y, 0.f);
        v.z = fmaxf(v.z, 0.f);
        v.w = fmaxf(v.w, 0.f);
        out4[i] = v;
    }
}

extern "C" void kernel_launch(void* const* d_in, const int* in_sizes, int n_in,
                              void* d_out, int out_size, void* d_ws, size_t ws_size,
                              hipStream_t stream) {
    const float* x = (const float*)d_in[0];
    const int*   h = (const int*)d_in[1];
    const int*   t = (const int*)d_in[2];
    float* out = (float*)d_out;

    const int n_nodes = in_sizes[0] / D_FEAT;   // 100000
    const int n_edges = in_sizes[1];            // 600000
    const int n4      = (n_nodes * D_FEAT) / 4; // out as float4

    // Workspace: [0, N) int deg ; [N, 2N) float dinv
    int*   deg  = (int*)d_ws;
    float* dinv = (float*)d_ws + n_nodes;

    {   // zero out + deg (covers n4 >> n_nodes)
        int blocks = (n4 + 255) / 256;
        gcn_zero_kernel<<<blocks, 256, 0, stream>>>((float4*)out, n4, deg, n_nodes);
    }
    {   // degree of source nodes
        int blocks = (n_edges + 255) / 256;
        gcn_degree_kernel<<<blocks, 256, 0, stream>>>(h, deg, n_edges);
    }
    {   // deg^-1/2
        int blocks = (n_nodes + 255) / 256;
        gcn_dinv_kernel<<<blocks, 256, 0, stream>>>(deg, dinv, n_nodes);
    }
    {   // async-gather + atomic scatter
        gcn_scatter_kernel<<<SCATTER_BLOCKS, 128, 0, stream>>>(x, h, t, dinv, out, n_edges);
    }
    {   // relu
        int blocks = (n4 + 255) / 256;
        gcn_relu_kernel<<<blocks, 256, 0, stream>>>((float4*)out, n4);
    }
}